// GraphOnly_36206574305755
// MI455X (gfx1250) — compile-verified
//
#include <hip/hip_runtime.h>
#include <math.h>

typedef __attribute__((ext_vector_type(2))) float v2f;
typedef __attribute__((ext_vector_type(4))) float v4f;
typedef __attribute__((ext_vector_type(8))) float v8f;

#define N_NODES 10000
#define IN_DIM  144
#define OUT_DIM 700
#define TH_DIM  256
#define BN_EPS  1e-5f

#define NPANEL  313            // ceil(10000/32) 32-column panels
#define NSTRIPE 8              // column-stripe parallelism
#define PPS     40             // panels per stripe (last stripe: 33)

// ---------------------------------------------------------------------------
// CDNA5 async global->LDS copy (ASYNCcnt-tracked), 16B per lane per instr
// ---------------------------------------------------------------------------
__device__ __forceinline__ void async_cp16(unsigned ldsAddr, const float* g) {
  asm volatile("global_load_async_to_lds_b128 %0, %1, off"
               :: "v"(ldsAddr), "v"(g) : "memory");
}
__device__ __forceinline__ void wait_async() {
  asm volatile("s_wait_asynccnt 0" ::: "memory");
}

// ---------------------------------------------------------------------------
// K1: per-feature mean / rstd over 10000 nodes (biased variance)
// ---------------------------------------------------------------------------
__global__ void bn_stats(const float* __restrict__ H, float* mean, float* rstd) {
  const int f = blockIdx.x;
  const int t = threadIdx.x;
  float s = 0.f, sq = 0.f;
  for (int i = t; i < N_NODES; i += 256) {
    float x = H[(size_t)i * IN_DIM + f];
    s += x; sq += x * x;
  }
  __shared__ float rs[256], rq[256];
  rs[t] = s; rq[t] = sq;
  __syncthreads();
  for (int o = 128; o > 0; o >>= 1) {
    if (t < o) { rs[t] += rs[t + o]; rq[t] += rq[t + o]; }
    __syncthreads();
  }
  if (t == 0) {
    float m = rs[0] / (float)N_NODES;
    float v = rq[0] / (float)N_NODES - m * m;
    mean[f] = m;
    rstd[f] = rsqrtf(v + BN_EPS);
  }
}

// ---------------------------------------------------------------------------
// K2: apply BN
// ---------------------------------------------------------------------------
__global__ void bn_apply(const float* __restrict__ H, const float* mean,
                         const float* rstd, const float* gamma,
                         const float* beta, float* Hn) {
  int i = blockIdx.x * blockDim.x + threadIdx.x;
  if (i >= N_NODES * IN_DIM) return;
  int f = i % IN_DIM;
  Hn[i] = (H[i] - mean[f]) * rstd[f] * gamma[f] + beta[f];
}

// ---------------------------------------------------------------------------
// K3: C[M,N] = A[M,K] @ W[K,N] + bias  via V_WMMA_F32_16X16X4_F32
//     one 16x16 tile per wave; unconditional clamped loads, select-to-zero.
// ---------------------------------------------------------------------------
__global__ __launch_bounds__(256) void gemm_bias(const float* __restrict__ A,
                                                 const float* __restrict__ W,
                                                 const float* __restrict__ bias,
                                                 float* __restrict__ C,
                                                 int M, int N, int K) {
  const int waveId = (blockIdx.x * blockDim.x + threadIdx.x) >> 5;
  const int lane   = threadIdx.x & 31;
  const int nT = (N + 15) >> 4;
  const int mT = M >> 4;
  if (waveId >= mT * nT) return;          // wave-uniform
  const int m0 = (waveId / nT) * 16;
  const int n0 = (waveId % nT) * 16;
  const int l15 = lane & 15;
  const int khi = (lane >> 4) * 2;
  const int row = m0 + l15;
  const int col = n0 + l15;
  const bool okc = (col < N);
  const int colc = okc ? col : (N - 1);   // clamp: loads always in-bounds

  v8f c = {};
#pragma unroll 2
  for (int k = 0; k < K; k += 4) {
    const int kk = k + khi;
    v2f a, b;
    a.x = A[(size_t)row * K + kk];
    a.y = A[(size_t)row * K + kk + 1];
    float bx = W[(size_t)kk * N + colc];
    float by = W[(size_t)(kk + 1) * N + colc];
    b.x = okc ? bx : 0.f;
    b.y = okc ? by : 0.f;
    c = __builtin_amdgcn_wmma_f32_16x16x4_f32(false, a, false, b,
                                              (short)0, c, false, false);
  }
  if (okc) {
    const float bv = bias[col];
    const int rbase = m0 + (lane >> 4) * 8;
#pragma unroll
    for (int r = 0; r < 8; ++r) {
      int rr = rbase + r;
      if (rr < M) C[(size_t)rr * N + col] = c[r] + bv;
    }
  }
}

// ---------------------------------------------------------------------------
// K4: in-place row L2-normalization of Hx[10000,256]  (one wave per row)
// ---------------------------------------------------------------------------
__global__ void row_normalize(float* __restrict__ Hx) {
  const int gw = (blockIdx.x * blockDim.x + threadIdx.x) >> 5;
  const int lane = threadIdx.x & 31;
  if (gw >= N_NODES) return;
  float* row = Hx + (size_t)gw * TH_DIM;
  float s = 0.f;
  float x[8];
#pragma unroll
  for (int u = 0; u < 8; ++u) { x[u] = row[lane + u * 32]; s += x[u] * x[u]; }
#pragma unroll
  for (int o = 16; o > 0; o >>= 1) s += __shfl_xor(s, o, 32);
  const float inv = 1.f / fmaxf(sqrtf(s), 1e-12f);
#pragma unroll
  for (int u = 0; u < 8; ++u) row[lane + u * 32] = x[u] * inv;
}

// ---------------------------------------------------------------------------
// branchless insert into descending-sorted top-5 (strict >, earlier wins ties)
// ---------------------------------------------------------------------------
__device__ __forceinline__ void ins5(float* tv, int* ti, float v, int j) {
  const bool c0 = v > tv[0], c1 = v > tv[1], c2 = v > tv[2],
             c3 = v > tv[3], c4 = v > tv[4];
  tv[4] = c4 ? (c3 ? tv[3] : v) : tv[4];  ti[4] = c4 ? (c3 ? ti[3] : j) : ti[4];
  tv[3] = c3 ? (c2 ? tv[2] : v) : tv[3];  ti[3] = c3 ? (c2 ? ti[2] : j) : ti[3];
  tv[2] = c2 ? (c1 ? tv[1] : v) : tv[2];  ti[2] = c2 ? (c1 ? ti[1] : j) : ti[2];
  tv[1] = c1 ? (c0 ? tv[0] : v) : tv[1];  ti[1] = c1 ? (c0 ? ti[0] : j) : ti[1];
  tv[0] = c0 ? v : tv[0];                 ti[0] = c0 ? j : ti[0];
}

// ---------------------------------------------------------------------------
// K5: fused cos = Xn @ Xn^T (f32 WMMA) + per-row top-5 (monotone => raw cos).
//     grid (79 m-groups, 8 column stripes); 8 waves/block, 16 rows/wave;
//     A rows and B panels staged via global_load_async_to_lds_b128.
// ---------------------------------------------------------------------------
__global__ __launch_bounds__(256) void cos_topk(const float* __restrict__ Xn,
                                                float* __restrict__ topvP,
                                                int* __restrict__ topiP) {
  __shared__ float sA[128 * TH_DIM];       // 128 KB: m-group's 128 rows
  __shared__ float sB[2][32 * TH_DIM];     // 2 x 32 KB: column panels

  const int t    = threadIdx.x;
  const int wave = t >> 5;
  const int lane = t & 31;
  const int l15  = lane & 15;
  const int khi  = (lane >> 4) * 2;
  const int mg   = blockIdx.x;
  const int cs   = blockIdx.y;
  const int m0   = (mg * 8 + wave) * 16;
  const bool waveActive = (m0 < N_NODES);

  const unsigned sAb  = (unsigned)(size_t)(void*)sA;
  const unsigned sBb0 = (unsigned)(size_t)(void*)sB[0];
  const unsigned sBb1 = (unsigned)(size_t)(void*)sB[1];

  const int pStart = cs * PPS;
  const int pEnd   = (pStart + PPS < NPANEL) ? (pStart + PPS) : NPANEL;

  // issue async loads for one 32-column panel into buffer bufSel
  auto issuePanel = [&](int p, int bufSel) {
    const int j0 = p * 32;
    const float* src = Xn + (size_t)j0 * TH_DIM;
    const unsigned dst = bufSel ? sBb1 : sBb0;
#pragma unroll
    for (int s = 0; s < 8; ++s) {
      const int fl = (s * 256 + t) * 4;     // float offset, 16B granules
      const int lc = fl >> 8;               // local column 0..31
      if (j0 + lc < N_NODES) async_cp16(dst + fl * 4, src + fl);
    }
  };

  // ---- stage this m-group's 128 A-rows + first panel (all async) ----
  issuePanel(pStart, 0);
  {
    const int rowBase = mg * 128;
    const float* srcA = Xn + (size_t)rowBase * TH_DIM;
#pragma unroll
    for (int u = 0; u < 32; ++u) {
      const int fl = (u * 256 + t) * 4;
      const int r  = fl >> 8;               // local row 0..127
      if (rowBase + r < N_NODES) async_cp16(sAb + fl * 4, srcA + fl);
    }
  }

  // ---- per-lane top-5 state for 8 rows ----
  float tv[8][5]; int ti[8][5];
#pragma unroll
  for (int r = 0; r < 8; ++r)
#pragma unroll
    for (int s = 0; s < 5; ++s) { tv[r][s] = -1e30f; ti[r][s] = 0; }

  wait_async();
  __syncthreads();

  for (int p = pStart; p < pEnd; ++p) {
    const int bl = (p - pStart) & 1;
    if (p + 1 < pEnd) issuePanel(p + 1, bl ^ 1);   // overlap with compute

    if (waveActive) {
      v8f c0 = {}, c1 = {};
      const float* Aw = sA + wave * 16 * TH_DIM;
      const float* Bb = sB[bl];
#pragma unroll 8
      for (int k = 0; k < TH_DIM; k += 4) {
        v2f a  = *(const v2f*)(Aw + l15 * TH_DIM + k + khi);
        v2f b0 = *(const v2f*)(Bb + l15 * TH_DIM + k + khi);
        v2f b1 = *(const v2f*)(Bb + (16 + l15) * TH_DIM + k + khi);
        c0 = __builtin_amdgcn_wmma_f32_16x16x4_f32(false, a, false, b0,
                                                   (short)0, c0, false, false);
        c1 = __builtin_amdgcn_wmma_f32_16x16x4_f32(false, a, false, b1,
                                                   (short)0, c1, false, false);
      }
      // ---- top-5 epilogue ----
      const int  j0c  = p * 32 + l15;                 // always < N_NODES
      const bool t1ok = (p * 32 + 16) < N_NODES;      // wave-uniform
      bool any = false;
#pragma unroll
      for (int r = 0; r < 8; ++r) {
        float mx = t1ok ? fmaxf(c0[r], c1[r]) : c0[r];
        any |= (mx > tv[r][4]);
      }
      if (__any((int)any)) {                          // rare after warm-up
#pragma unroll
        for (int r = 0; r < 8; ++r) ins5(tv[r], ti[r], c0[r], j0c);
        if (t1ok) {
#pragma unroll
          for (int r = 0; r < 8; ++r) ins5(tv[r], ti[r], c1[r], j0c + 16);
        }
      }
    }
    wait_async();          // own async writes (next panel) landed in LDS
    __syncthreads();       // cross-wave visibility before buffer swap
  }

  // ---- cross-lane merge: reuse sA; 2560 floats per wave ----
  __syncthreads();
  if (waveActive) {
    float* mv = sA + wave * 2560;
#pragma unroll
    for (int r = 0; r < 8; ++r) {
      const int rr = r + (lane >> 4) * 8;
#pragma unroll
      for (int s = 0; s < 5; ++s) {
        const int base = ((rr * 16 + l15) * 5 + s) * 2;
        mv[base] = tv[r][s];
        ((int*)mv)[base + 1] = ti[r][s];
      }
    }
  }
  __syncthreads();
  if (waveActive && lane < 16) {
    const float* mv = sA + wave * 2560;
    float bv[5]; int bi[5];
#pragma unroll
    for (int s = 0; s < 5; ++s) { bv[s] = -1e30f; bi[s] = 0; }
    for (int src = 0; src < 16; ++src)
#pragma unroll
      for (int s = 0; s < 5; ++s) {
        const int base = ((lane * 16 + src) * 5 + s) * 2;
        ins5(bv, bi, mv[base], ((const int*)mv)[base + 1]);
      }
    const int row = m0 + lane;
#pragma unroll
    for (int s = 0; s < 5; ++s) {
      topvP[((size_t)cs * N_NODES + row) * 5 + s] = bv[s];
      topiP[((size_t)cs * N_NODES + row) * 5 + s] = bi[s];
    }
  }
}

// ---------------------------------------------------------------------------
// K6: merge the NSTRIPE partial top-5 lists per row (stripe order = ascending
//     j, so strict-> insertion preserves the reference tie semantics)
// ---------------------------------------------------------------------------
__global__ void merge_topk(const float* __restrict__ topvP,
                           const int* __restrict__ topiP,
                           float* __restrict__ topv, int* __restrict__ topi) {
  int i = blockIdx.x * blockDim.x + threadIdx.x;
  if (i >= N_NODES) return;
  float bv[5]; int bi[5];
#pragma unroll
  for (int s = 0; s < 5; ++s) { bv[s] = -1e30f; bi[s] = 0; }
  for (int cs = 0; cs < NSTRIPE; ++cs)
#pragma unroll
    for (int s = 0; s < 5; ++s) {
      size_t o = ((size_t)cs * N_NODES + i) * 5 + s;
      ins5(bv, bi, topvP[o], topiP[o]);
    }
#pragma unroll
  for (int s = 0; s < 5; ++s) { topv[i * 5 + s] = bv[s]; topi[i * 5 + s] = bi[s]; }
}

// ---------------------------------------------------------------------------
// K7: transform selected cosines -> A values, row-sum -> d^-1/2
// ---------------------------------------------------------------------------
__global__ void transform_a(const float* __restrict__ topv, float* __restrict__ Aval,
                            float* __restrict__ drs) {
  int i = blockIdx.x * blockDim.x + threadIdx.x;
  if (i >= N_NODES) return;
  float s = 0.f;
#pragma unroll
  for (int k = 0; k < 5; ++k) {
    float cz  = fminf(fmaxf(topv[i * 5 + k], -1.f), 1.f);
    float sam = acosf(cz);
    float e   = expf(-0.2f * sam);
    float a   = 1.f / (1.f + expf(-e));     // sigmoid
    a = fmaxf(a, 0.1f);                     // clip(.,0.1,None)
    Aval[i * 5 + k] = a;
    s += a;
  }
  drs[i] = rsqrtf(s);
}

// ---------------------------------------------------------------------------
// K8: out[i,:] = leakyrelu( sum_k Ahat(i,jk) * HW[jk,:] )
// ---------------------------------------------------------------------------
__global__ void gather_out(const float* __restrict__ Aval, const int* __restrict__ topi,
                           const float* __restrict__ drs, const float* __restrict__ HW,
                           float* __restrict__ out) {
  const int i = blockIdx.x;
  __shared__ float w[5]; __shared__ int jj[5];
  if (threadIdx.x < 5) {
    const int k = threadIdx.x;
    const int j = topi[i * 5 + k];
    jj[k] = j;
    w[k]  = Aval[i * 5 + k] * drs[i] * drs[j];
  }
  __syncthreads();
  for (int c = threadIdx.x; c < OUT_DIM; c += blockDim.x) {
    float acc = 0.f;
#pragma unroll
    for (int k = 0; k < 5; ++k) acc += w[k] * HW[(size_t)jj[k] * OUT_DIM + c];
    out[(size_t)i * OUT_DIM + c] = (acc >= 0.f) ? acc : 0.01f * acc;
  }
}

// ---------------------------------------------------------------------------
// K9: scatter masked-A nonzeros into the (pre-zeroed) dense A output
// ---------------------------------------------------------------------------
__global__ void scatter_a(const float* __restrict__ Aval, const int* __restrict__ topi,
                          float* __restrict__ Aout) {
  int i = blockIdx.x * blockDim.x + threadIdx.x;
  if (i >= N_NODES) return;
#pragma unroll
  for (int k = 0; k < 5; ++k)
    Aout[(size_t)i * N_NODES + topi[i * 5 + k]] = Aval[i * 5 + k];
}

// ---------------------------------------------------------------------------
extern "C" void kernel_launch(void* const* d_in, const int* in_sizes, int n_in,
                              void* d_out, int out_size, void* d_ws, size_t ws_size,
                              hipStream_t stream) {
  const float* H     = (const float*)d_in[0];
  const float* gamma = (const float*)d_in[1];
  const float* beta  = (const float*)d_in[2];
  const float* Wt    = (const float*)d_in[3];
  const float* bt    = (const float*)d_in[4];
  const float* Wo    = (const float*)d_in[5];
  const float* bo    = (const float*)d_in[6];

  // workspace layout (floats), ~20 MB total
  float* ws    = (float*)d_ws;
  float* mean  = ws;                                      // 144
  float* rstd  = ws + 256;                                // 144
  float* Hn    = ws + 512;                                // 1,440,000
  float* Xn    = Hn + (size_t)N_NODES * IN_DIM;           // 2,560,000
  float* topvP = Xn + (size_t)N_NODES * TH_DIM;           // 8*50,000
  int*   topiP = (int*)(topvP + (size_t)NSTRIPE * 5 * N_NODES);   // 8*50,000
  float* topv  = (float*)(topiP + (size_t)NSTRIPE * 5 * N_NODES); // 50,000
  int*   topi  = (int*)(topv + 5 * N_NODES);              // 50,000
  float* Aval  = (float*)(topi + 5 * N_NODES);            // 50,000
  float* drs   = Aval + 5 * N_NODES;                      // 10,000

  float* out   = (float*)d_out;                           // [10000,700]
  float* Aout  = out + (size_t)N_NODES * OUT_DIM;         // [10000,10000]
  float* HW    = Aout;   // park Hn@W_out+b_out (28 MB) in A region; zeroed later

  bn_stats<<<IN_DIM, 256, 0, stream>>>(H, mean, rstd);
  bn_apply<<<(N_NODES * IN_DIM + 255) / 256, 256, 0, stream>>>(H, mean, rstd,
                                                               gamma, beta, Hn);
  gemm_bias<<<(625 * 16) / 8, 256, 0, stream>>>(Hn, Wt, bt, Xn,
                                                N_NODES, TH_DIM, IN_DIM);
  row_normalize<<<(N_NODES * 32 + 255) / 256, 256, 0, stream>>>(Xn);
  gemm_bias<<<(625 * 44 + 7) / 8, 256, 0, stream>>>(Hn, Wo, bo, HW,
                                                    N_NODES, OUT_DIM, IN_DIM);
  cos_topk<<<dim3(79, NSTRIPE), 256, 0, stream>>>(Xn, topvP, topiP);
  merge_topk<<<(N_NODES + 255) / 256, 256, 0, stream>>>(topvP, topiP, topv, topi);
  transform_a<<<(N_NODES + 255) / 256, 256, 0, stream>>>(topv, Aval, drs);
  gather_out<<<N_NODES, 256, 0, stream>>>(Aval, topi, drs, HW, out);
  hipMemsetAsync(Aout, 0, (size_t)N_NODES * N_NODES * sizeof(float), stream);
  scatter_a<<<(N_NODES + 255) / 256, 256, 0, stream>>>(Aval, topi, Aout);
}